// DMoN_DPR_10677288698440
// MI455X (gfx1250) — compile-verified
//
#include <hip/hip_runtime.h>
#include <hip/hip_bf16.h>

typedef __attribute__((ext_vector_type(2))) float v2f;
typedef __attribute__((ext_vector_type(8))) float v8f;

// ---------------------------------------------------------------------------
// SELU
// ---------------------------------------------------------------------------
__device__ __forceinline__ float selu_f(float x) {
    const float scale = 1.0507009873554804934193349852946f;
    const float alpha = 1.6732632423543772848170429916717f;
    return scale * (x > 0.0f ? x : alpha * (__expf(x) - 1.0f));
}

// ---------------------------------------------------------------------------
// Degree / normalization kernels
// ---------------------------------------------------------------------------
__global__ void deg_init_kernel(float* __restrict__ deg, long long n) {
    long long i = blockIdx.x * (long long)blockDim.x + threadIdx.x;
    if (i < n) deg[i] = 1.0f;  // self-loop weight
}

__global__ void deg_scatter_kernel(const long long* __restrict__ col,
                                   const float* __restrict__ w,
                                   float* __restrict__ deg, long long E) {
    long long e = blockIdx.x * (long long)blockDim.x + threadIdx.x;
    if (e < E) atomicAdd(&deg[col[e]], w[e]);
}

__global__ void dinv_kernel(const float* __restrict__ deg,
                            float* __restrict__ dinv, long long n) {
    long long i = blockIdx.x * (long long)blockDim.x + threadIdx.x;
    if (i < n) {
        float d = deg[i];
        dinv[i] = d > 0.0f ? rsqrtf(d) : 0.0f;
    }
}

// ---------------------------------------------------------------------------
// Dense GEMM via native fp32 WMMA with LDS-staged B panel.
//   C[M,N] = A[M,K] @ B[K,N] (+ bias)
// Block = 2 waves (Mtiles=6250 even -> exact grid, EXEC all-1s at WMMA).
// Each wave computes a 16 x (16*NT) tile: NT v8f accumulators, A fragment
// reused across NT WMMAs.
// B staged in LDS in k-PAIRED layout sB[k/2][n][k%2] so each WMMA B-fragment
// (B[k][n], B[k+1][n]) is one contiguous 8B ds_load_b64 into an even VGPR
// pair -- no register-packing movs between DS load and v_wmma.
// M multiple of 16, N multiple of 16, K multiple of 32.
// ---------------------------------------------------------------------------
template <int NT>
__global__ void __launch_bounds__(64)
gemm_wmma_f32_tiled(const float* __restrict__ A, const float* __restrict__ B,
                    const float* __restrict__ bias, float* __restrict__ C,
                    int M, int K, int N) {
    constexpr int KC = 32;                 // K-chunk staged in LDS
    constexpr int NW = 16 * NT;            // tile width
    __shared__ float sB[KC / 2][NW][2];    // k-paired; NT=4 -> 8 KB

    const int tid  = threadIdx.x;
    const int wave = tid >> 5;
    const int lane = tid & 31;
    const int lo   = lane & 15;            // m (A) / n (B,C,D)
    const int hi   = lane >> 4;            // K-pair select 0/1 vs 2/3

    const int mtile = blockIdx.x * 2 + wave;
    const int nbase = blockIdx.y * NW;
    const bool active = (mtile * 16 < M);

    const int m = (active ? mtile * 16 + lo : lo);
    const float* __restrict__ arow = A + (size_t)m * K + 2 * hi;

    v8f acc[NT];
#pragma unroll
    for (int t = 0; t < NT; ++t) acc[t] = (v8f){};

    for (int k0 = 0; k0 < K; k0 += KC) {
        // Cooperative fill: float4 global load, scatter into paired layout
        constexpr int F4_PER_ROW = NW / 4;
        constexpr int F4_TOTAL   = KC * F4_PER_ROW;
        for (int idx = tid; idx < F4_TOTAL; idx += 64) {
            int kk = idx / F4_PER_ROW;
            int n0 = (idx % F4_PER_ROW) * 4;
            float4 v = *((const float4*)(B + (size_t)(k0 + kk) * N + nbase) +
                         (n0 >> 2));
            int kp = kk >> 1, ko = kk & 1;
            sB[kp][n0 + 0][ko] = v.x;
            sB[kp][n0 + 1][ko] = v.y;
            sB[kp][n0 + 2][ko] = v.z;
            sB[kp][n0 + 3][ko] = v.w;
        }
        __syncthreads();

        if (active) {
            // prefetch next A chunk (speculative; dropped if OOB)
            __builtin_prefetch(arow + k0 + KC, 0, 0);
#pragma unroll
            for (int kk = 0; kk < KC; kk += 4) {
                v2f a;  // contiguous pair -> single b64 load
                a.x = arow[k0 + kk];
                a.y = arow[k0 + kk + 1];
#pragma unroll
                for (int t = 0; t < NT; ++t) {
                    // (B[k][n], B[k+1][n]) contiguous: one ds_load_b64
                    v2f b = *(const v2f*)&sB[(kk + 2 * hi) >> 1][t * 16 + lo][0];
                    acc[t] = __builtin_amdgcn_wmma_f32_16x16x4_f32(
                        false, a, false, b, (short)0, acc[t], false, false);
                }
            }
        }
        __syncthreads();
    }

    if (active) {
#pragma unroll
        for (int t = 0; t < NT; ++t) {
            const int n  = nbase + t * 16 + lo;
            const float bv = bias ? bias[n] : 0.0f;
#pragma unroll
            for (int j = 0; j < 8; ++j) {
                // D layout: VGPR j, lanes 0-15 -> row j, lanes 16-31 -> row j+8
                const int row = mtile * 16 + j + 8 * hi;
                C[(size_t)row * N + n] = acc[t][j] + bv;
            }
        }
    }
}

// ---------------------------------------------------------------------------
// agg[i,:] = dinv[i]^2 * hW[i,:]   (self-loop term; also initializes agg)
// ---------------------------------------------------------------------------
__global__ void selfloop_init_kernel(const float* __restrict__ hW,
                                     const float* __restrict__ dinv,
                                     float* __restrict__ agg,
                                     int fshift, long long total) {
    long long idx = blockIdx.x * (long long)blockDim.x + threadIdx.x;
    if (idx >= total) return;
    long long i = idx >> fshift;
    float di = dinv[i];
    agg[idx] = di * di * hW[idx];
}

// ---------------------------------------------------------------------------
// Edge scatter, wave-per-edge: agg[col,:] += dinv[row]*w*dinv[col]*hW[row,:]
// Edge metadata loaded once per wave; lanes stride features (coalesced loads
// + f32 atomics, L2-resident since the hW panel <= 51 MB << 192 MB L2).
// ---------------------------------------------------------------------------
__global__ void scatter_edges_wave_kernel(const long long* __restrict__ rowi,
                                          const long long* __restrict__ coli,
                                          const float* __restrict__ w,
                                          const float* __restrict__ dinv,
                                          const float* __restrict__ hW,
                                          float* __restrict__ agg,
                                          int Nf, long long E) {
    long long wid = blockIdx.x * (long long)(blockDim.x >> 5) + (threadIdx.x >> 5);
    int lane = threadIdx.x & 31;
    if (wid >= E) return;
    long long r = rowi[wid];
    long long c = coli[wid];
    float nw = dinv[r] * w[wid] * dinv[c];
    const float* __restrict__ src = hW + (size_t)r * Nf;
    float* dst = agg + (size_t)c * Nf;
    for (int f = lane; f < Nf; f += 32) {
        atomicAdd(dst + f, nw * src[f]);
    }
}

// ---------------------------------------------------------------------------
// h = selu(agg + bias[f] + skip)
// ---------------------------------------------------------------------------
__global__ void epilogue_selu_kernel(const float* __restrict__ agg,
                                     const float* __restrict__ bias,
                                     const float* __restrict__ skip,
                                     float* __restrict__ out,
                                     int fmask, long long total) {
    long long idx = blockIdx.x * (long long)blockDim.x + threadIdx.x;
    if (idx >= total) return;
    int f = (int)(idx & fmask);
    float v = agg[idx] + bias[f] + skip[idx];
    out[idx] = selu_f(v);
}

// ---------------------------------------------------------------------------
// Final: out[i,:] = softmax(selu(agg[i,:] + b2))  over K columns (K<=32)
// ---------------------------------------------------------------------------
__global__ void final_softmax_kernel(const float* __restrict__ agg,
                                     const float* __restrict__ bias,
                                     float* __restrict__ out,
                                     int Kc, long long n) {
    long long i = blockIdx.x * (long long)blockDim.x + threadIdx.x;
    if (i >= n) return;
    float vals[32];
    float mx = -3.4e38f;
    for (int f = 0; f < Kc; ++f) {
        float v = selu_f(agg[i * Kc + f] + bias[f]);
        vals[f] = v;
        mx = fmaxf(mx, v);
    }
    float s = 0.0f;
    for (int f = 0; f < Kc; ++f) {
        float e = __expf(vals[f] - mx);
        vals[f] = e;
        s += e;
    }
    float inv = 1.0f / s;
    for (int f = 0; f < Kc; ++f) out[i * Kc + f] = vals[f] * inv;
}

// ---------------------------------------------------------------------------
// Host launch
// ---------------------------------------------------------------------------
static inline unsigned gridFor(long long total, int block) {
    return (unsigned)((total + block - 1) / block);
}

static void launch_gemm(const float* A, const float* B, const float* bias,
                        float* C, int M, int K, int N, hipStream_t stream) {
    int mtiles = M / 16;
    int gx = (mtiles + 1) / 2;  // 2 waves per block
    if (N % 64 == 0) {
        dim3 grid(gx, N / 64);
        gemm_wmma_f32_tiled<4><<<grid, 64, 0, stream>>>(A, B, bias, C, M, K, N);
    } else {
        dim3 grid(gx, N / 16);
        gemm_wmma_f32_tiled<1><<<grid, 64, 0, stream>>>(A, B, bias, C, M, K, N);
    }
}

static int ilog2i(int v) { int l = 0; while ((1 << l) < v) ++l; return l; }

extern "C" void kernel_launch(void* const* d_in, const int* in_sizes, int n_in,
                              void* d_out, int out_size, void* d_ws, size_t ws_size,
                              hipStream_t stream) {
    const float*     x    = (const float*)d_in[0];
    const long long* ei   = (const long long*)d_in[1];   // int64 (2, E)
    const float*     ew   = (const float*)d_in[2];
    const float*     W0   = (const float*)d_in[3];
    const float*     b0   = (const float*)d_in[4];
    const float*     P0w  = (const float*)d_in[5];
    const float*     P0b  = (const float*)d_in[6];
    const float*     W1   = (const float*)d_in[7];
    const float*     b1   = (const float*)d_in[8];
    const float*     P1w  = (const float*)d_in[9];
    const float*     P1b  = (const float*)d_in[10];
    const float*     W2   = (const float*)d_in[11];
    const float*     b2   = (const float*)d_in[12];
    float*           out  = (float*)d_out;

    const int HID  = in_sizes[4];             // 128
    const int HID2 = in_sizes[8];             // 64
    const int Kc   = in_sizes[12];            // 16
    const int IN   = in_sizes[3] / HID;       // 256
    const long long Nn = (long long)in_sizes[0] / IN;   // 100000
    const long long E  = (long long)in_sizes[2];        // 3200000

    const long long* rowi = ei;       // edge_index[0]
    const long long* coli = ei + E;   // edge_index[1]

    // Workspace layout (floats)
    float* hWbuf   = (float*)d_ws;               // Nn*HID
    float* skipbuf = hWbuf + Nn * HID;           // Nn*HID
    float* aggbuf  = skipbuf + Nn * HID;         // Nn*HID
    float* deg     = aggbuf + Nn * HID;          // Nn
    float* dinv    = deg + Nn;                   // Nn
    (void)ws_size; (void)n_in; (void)out_size;

    const int BLK = 256;
    const int WPB = BLK / 32;  // waves per block for edge scatter

    // --- gcn_norm: deg -> dinv ---
    deg_init_kernel<<<gridFor(Nn, BLK), BLK, 0, stream>>>(deg, Nn);
    deg_scatter_kernel<<<gridFor(E, BLK), BLK, 0, stream>>>(coli, ew, deg, E);
    dinv_kernel<<<gridFor(Nn, BLK), BLK, 0, stream>>>(deg, dinv, Nn);

    // ======================= Layer 0 (IN -> HID) =======================
    launch_gemm(x, W0,  nullptr, hWbuf,   (int)Nn, IN, HID, stream);   // hW0
    launch_gemm(x, P0w, P0b,     skipbuf, (int)Nn, IN, HID, stream);   // skip0

    {
        long long total = Nn * HID;
        selfloop_init_kernel<<<gridFor(total, BLK), BLK, 0, stream>>>(
            hWbuf, dinv, aggbuf, ilog2i(HID), total);
        scatter_edges_wave_kernel<<<gridFor(E * WPB, BLK) , BLK, 0, stream>>>(
            rowi, coli, ew, dinv, hWbuf, aggbuf, HID, E);
        // h0 = selu(agg + b0 + skip0) -> overwrite skipbuf (elementwise, safe)
        epilogue_selu_kernel<<<gridFor(total, BLK), BLK, 0, stream>>>(
            aggbuf, b0, skipbuf, skipbuf, HID - 1, total);
    }
    float* h0 = skipbuf;

    // ======================= Layer 1 (HID -> HID2) =====================
    float* hW1   = hWbuf;                 // Nn*HID2
    float* agg1  = aggbuf;                // Nn*HID2
    float* skip1 = aggbuf + Nn * HID2;    // Nn*HID2 (second half of agg panel)

    launch_gemm(h0, W1,  nullptr, hW1,   (int)Nn, HID, HID2, stream);
    launch_gemm(h0, P1w, P1b,     skip1, (int)Nn, HID, HID2, stream);

    {
        long long total = Nn * HID2;
        selfloop_init_kernel<<<gridFor(total, BLK), BLK, 0, stream>>>(
            hW1, dinv, agg1, ilog2i(HID2), total);
        scatter_edges_wave_kernel<<<gridFor(E * WPB, BLK), BLK, 0, stream>>>(
            rowi, coli, ew, dinv, hW1, agg1, HID2, E);
        // h1 = selu(agg1 + b1 + skip1) -> write into skipbuf (h0 is dead)
        epilogue_selu_kernel<<<gridFor(total, BLK), BLK, 0, stream>>>(
            agg1, b1, skip1, skipbuf, HID2 - 1, total);
    }
    float* h1 = skipbuf;

    // ======================= Layer 2 (HID2 -> K) =======================
    float* hW2  = hWbuf;    // Nn*Kc
    float* agg2 = aggbuf;   // Nn*Kc

    launch_gemm(h1, W2, nullptr, hW2, (int)Nn, HID2, Kc, stream);

    {
        long long total = Nn * Kc;
        selfloop_init_kernel<<<gridFor(total, BLK), BLK, 0, stream>>>(
            hW2, dinv, agg2, ilog2i(Kc), total);
        scatter_edges_wave_kernel<<<gridFor(E * WPB, BLK), BLK, 0, stream>>>(
            rowi, coli, ew, dinv, hW2, agg2, Kc, E);
        final_softmax_kernel<<<gridFor(Nn, BLK), BLK, 0, stream>>>(
            agg2, b2, out, Kc, Nn);
    }
}